// GCN_39307540693086
// MI455X (gfx1250) — compile-verified
//
#include <hip/hip_runtime.h>

#define N_NODES  100000
#define N_EDGES  3200000
#define N_GRAPHS 1024
#define DIM      64

typedef __attribute__((ext_vector_type(2))) float v2f;
typedef __attribute__((ext_vector_type(8))) float v8f;

// ---------------- degree / norm ----------------
__global__ void deg_init_k(float* deg, int n) {
    int i = blockIdx.x * blockDim.x + threadIdx.x;
    if (i < n) deg[i] = 1.0f;                 // self-loop contributes 1
}
__global__ void deg_edge_k(const int* __restrict__ col, float* deg, int e) {
    int i = blockIdx.x * blockDim.x + threadIdx.x;
    if (i < e) atomicAdd(&deg[col[i]], 1.0f);
}
__global__ void dinv_k(float* deg, int n) {
    int i = blockIdx.x * blockDim.x + threadIdx.x;
    if (i < n) deg[i] = rsqrtf(deg[i]);       // deg >= 1 always
}
__global__ void norm_k(const int* __restrict__ row, const int* __restrict__ col,
                       const float* __restrict__ dinv, float* __restrict__ norm, int e) {
    int i = blockIdx.x * blockDim.x + threadIdx.x;
    if (i < e) norm[i] = dinv[row[i]] * dinv[col[i]];
}

// ---------------- input transform: out = relu(x @ W0 + b0) ----------------
__global__ __launch_bounds__(64) void xform_k(const float* __restrict__ x,
                                              const float* __restrict__ W0,
                                              const float* __restrict__ b0,
                                              float* __restrict__ out) {
    __shared__ float xs[15];
    int i = blockIdx.x, t = threadIdx.x;
    if (t < 15) xs[t] = x[(size_t)i * 15 + t];
    __syncthreads();
    float s = b0[t];
#pragma unroll
    for (int j = 0; j < 15; ++j) s += xs[j] * W0[j * 64 + t];
    out[(size_t)i * 64 + t] = fmaxf(s, 0.0f);
}

// ---------------- WMMA f32 GEMM: dst[N,64] = src[N,64] @ Wc[64,64] ----------------
__global__ __launch_bounds__(256) void gemm_wc_k(const float* __restrict__ src,
                                                 const float* __restrict__ Wc,
                                                 float* __restrict__ dst, int ntiles) {
    int lane = threadIdx.x & 31;
    int tile = blockIdx.x * 8 + (threadIdx.x >> 5);
    if (tile >= ntiles) return;               // whole-wave uniform exit: EXEC stays all-1s
    int row0 = tile * 16;
    int m  = lane & 15;
    int kh = (lane >> 4) * 2;                 // K sub-offset per lane half (ISA 16x4 f32 layout)
    const float* arow = src + (size_t)(row0 + m) * 64;
    v8f acc0 = {}, acc1 = {}, acc2 = {}, acc3 = {};
#pragma unroll
    for (int k = 0; k < 16; ++k) {
        int kb = k * 4 + kh;
        v2f a; a.x = arow[kb]; a.y = arow[kb + 1];
        const float* br0 = Wc + (size_t)kb * 64;
        const float* br1 = br0 + 64;
        v2f b0; b0.x = br0[m];      b0.y = br1[m];
        v2f b1; b1.x = br0[m + 16]; b1.y = br1[m + 16];
        v2f b2; b2.x = br0[m + 32]; b2.y = br1[m + 32];
        v2f b3; b3.x = br0[m + 48]; b3.y = br1[m + 48];
        acc0 = __builtin_amdgcn_wmma_f32_16x16x4_f32(false, a, false, b0, (short)0, acc0, false, false);
        acc1 = __builtin_amdgcn_wmma_f32_16x16x4_f32(false, a, false, b1, (short)0, acc1, false, false);
        acc2 = __builtin_amdgcn_wmma_f32_16x16x4_f32(false, a, false, b2, (short)0, acc2, false, false);
        acc3 = __builtin_amdgcn_wmma_f32_16x16x4_f32(false, a, false, b3, (short)0, acc3, false, false);
    }
    int mh = (lane >> 4) * 8;                 // D layout: VGPR v -> M=v (lanes 0-15), M=v+8 (16-31)
#pragma unroll
    for (int v = 0; v < 8; ++v) {
        float* drow = dst + (size_t)(row0 + mh + v) * 64;
        drow[m]      = acc0[v];
        drow[16 + m] = acc1[v];
        drow[32 + m] = acc2[v];
        drow[48 + m] = acc3[v];
    }
}

// ---------------- propagation: self-loop init, edge scatter, relu+bias ----------------
__global__ void agg_init_k(const float* __restrict__ h, const float* __restrict__ dinv,
                           float* __restrict__ agg, int total) {
    int idx = blockIdx.x * blockDim.x + threadIdx.x;
    if (idx >= total) return;
    int i = idx >> 6;
    float dv = dinv[i];
    agg[idx] = h[idx] * dv * dv;              // self-loop message, norm = dinv^2
}
__global__ __launch_bounds__(256) void edge_msg_k(const int* __restrict__ row,
                                                  const int* __restrict__ col,
                                                  const float* __restrict__ norm,
                                                  const float* __restrict__ h,
                                                  float* __restrict__ agg, int e) {
    int lane = threadIdx.x & 31;
    int edge = blockIdx.x * 8 + (threadIdx.x >> 5);
    if (edge >= e) return;
    int r = row[edge], c = col[edge];
    float nv = norm[edge];
    const float2* hp = (const float2*)(h + (size_t)r * 64);
    float2 v = hp[lane];
    float* ap = agg + (size_t)c * 64 + lane * 2;
    atomicAdd(ap,     v.x * nv);
    atomicAdd(ap + 1, v.y * nv);
}
__global__ void relu_bias_k(const float* __restrict__ agg, const float* __restrict__ bc,
                            float* __restrict__ out, int total) {
    int idx = blockIdx.x * blockDim.x + threadIdx.x;
    if (idx >= total) return;
    out[idx] = fmaxf(agg[idx] + bc[idx & 63], 0.0f);
}

// ---------------- Set2Set ----------------
__device__ __forceinline__ unsigned int fkey(float x) {
    unsigned int u = __float_as_uint(x);
    return (u & 0x80000000u) ? ~u : (u | 0x80000000u);
}
__device__ __forceinline__ float unfkey(unsigned int u) {
    return __uint_as_float((u & 0x80000000u) ? (u & 0x7FFFFFFFu) : ~u);
}

__global__ void zero_f_k(float* p, int n) {
    int i = blockIdx.x * blockDim.x + threadIdx.x;
    if (i < n) p[i] = 0.0f;
}

__global__ __launch_bounds__(256) void lstm_k(const float* __restrict__ q_star,
                                              float* __restrict__ h_l, float* __restrict__ c_l,
                                              const float* __restrict__ Wih,
                                              const float* __restrict__ Whh,
                                              const float* __restrict__ bih,
                                              const float* __restrict__ bhh) {
    __shared__ float qs[128], hs[64], gs[256];
    int b = blockIdx.x, t = threadIdx.x;
    if (t < 128) qs[t] = q_star[(size_t)b * 128 + t];
    else if (t < 192) hs[t - 128] = h_l[(size_t)b * 64 + (t - 128)];
    __syncthreads();
    float s = bih[t] + bhh[t];
    const float* wi = Wih + (size_t)t * 128;
    const float* wh = Whh + (size_t)t * 64;
#pragma unroll 4
    for (int j = 0; j < 128; ++j) s += wi[j] * qs[j];
#pragma unroll 4
    for (int j = 0; j < 64; ++j) s += wh[j] * hs[j];
    gs[t] = s;
    __syncthreads();
    if (t < 64) {
        float ig = 1.0f / (1.0f + expf(-gs[t]));
        float fg = 1.0f / (1.0f + expf(-gs[64 + t]));
        float gg = tanhf(gs[128 + t]);
        float og = 1.0f / (1.0f + expf(-gs[192 + t]));
        float c = fg * c_l[(size_t)b * 64 + t] + ig * gg;
        c_l[(size_t)b * 64 + t] = c;
        h_l[(size_t)b * 64 + t] = og * tanhf(c);
    }
}

__global__ __launch_bounds__(64) void att_init_k(unsigned int* m_u, float* denom, float* r) {
    int b = blockIdx.x, t = threadIdx.x;
    if (t == 0) { m_u[b] = 0u; denom[b] = 0.0f; }   // key 0 == -NaN sentinel (minimum)
    r[(size_t)b * 64 + t] = 0.0f;
}

__global__ __launch_bounds__(256) void e_max_k(const float* __restrict__ out,
                                               const float* __restrict__ h_l,
                                               const int* __restrict__ batch,
                                               float* __restrict__ e_arr,
                                               unsigned int* __restrict__ m_u, int n) {
    int lane = threadIdx.x & 31;
    int node = blockIdx.x * 8 + (threadIdx.x >> 5);
    if (node >= n) return;
    int g = batch[node];
    const float2* op = (const float2*)(out + (size_t)node * 64);
    const float2* qp = (const float2*)(h_l + (size_t)g * 64);
    float2 a = op[lane], q = qp[lane];
    float s = a.x * q.x + a.y * q.y;
    for (int o = 16; o > 0; o >>= 1) s += __shfl_xor(s, o, 32);
    if (lane == 0) { e_arr[node] = s; atomicMax(&m_u[g], fkey(s)); }
}

__global__ void anum_k(const float* __restrict__ e_arr, const int* __restrict__ batch,
                       const unsigned int* __restrict__ m_u,
                       float* __restrict__ a_num, float* __restrict__ denom, int n) {
    int i = blockIdx.x * blockDim.x + threadIdx.x;
    if (i >= n) return;
    int g = batch[i];
    float v = expf(e_arr[i] - unfkey(m_u[g]));
    a_num[i] = v;
    atomicAdd(&denom[g], v);
}

__global__ __launch_bounds__(256) void r_accum_k(const float* __restrict__ out,
                                                 const float* __restrict__ a_num,
                                                 const float* __restrict__ denom,
                                                 const int* __restrict__ batch,
                                                 float* __restrict__ r, int n) {
    int lane = threadIdx.x & 31;
    int node = blockIdx.x * 8 + (threadIdx.x >> 5);
    if (node >= n) return;
    int g = batch[node];
    float coeff = a_num[node] / denom[g];
    const float2* op = (const float2*)(out + (size_t)node * 64);
    float2 v = op[lane];
    float* rp = r + (size_t)g * 64 + lane * 2;
    atomicAdd(rp,     v.x * coeff);
    atomicAdd(rp + 1, v.y * coeff);
}

__global__ __launch_bounds__(64) void qstar_k(const float* __restrict__ h_l,
                                              const float* __restrict__ r,
                                              float* __restrict__ q_star) {
    int b = blockIdx.x, t = threadIdx.x;
    q_star[(size_t)b * 128 + t]      = h_l[(size_t)b * 64 + t];
    q_star[(size_t)b * 128 + 64 + t] = r[(size_t)b * 64 + t];
}

// ---------------- final MLP ----------------
__global__ __launch_bounds__(64) void mlp_k(const float* __restrict__ q_star,
                                            const float* __restrict__ W1, const float* __restrict__ b1,
                                            const float* __restrict__ W2, const float* __restrict__ b2,
                                            float* __restrict__ y) {
    __shared__ float qs[128], ys[64];
    int b = blockIdx.x, t = threadIdx.x;
    qs[t]      = q_star[(size_t)b * 128 + t];
    qs[64 + t] = q_star[(size_t)b * 128 + 64 + t];
    __syncthreads();
    float s = b1[t];
#pragma unroll 4
    for (int j = 0; j < 128; ++j) s += qs[j] * W1[j * 64 + t];
    ys[t] = fmaxf(s, 0.0f);
    __syncthreads();
    if (t < 12) {
        float o = b2[t];
#pragma unroll 4
        for (int j = 0; j < 64; ++j) o += ys[j] * W2[j * 12 + t];
        y[(size_t)b * 12 + t] = o;
    }
}

// ---------------- launcher ----------------
extern "C" void kernel_launch(void* const* d_in, const int* in_sizes, int n_in,
                              void* d_out, int out_size, void* d_ws, size_t ws_size,
                              hipStream_t stream) {
    const int N = N_NODES, E = N_EDGES, B = N_GRAPHS;

    const float* x    = (const float*)d_in[0];
    const int*   ei   = (const int*)d_in[1];
    const int*   bat  = (const int*)d_in[2];
    const float* W0   = (const float*)d_in[3];
    const float* b0   = (const float*)d_in[4];
    const float* Wc   = (const float*)d_in[5];
    const float* bc   = (const float*)d_in[6];
    const float* Wih  = (const float*)d_in[7];
    const float* Whh  = (const float*)d_in[8];
    const float* bih  = (const float*)d_in[9];
    const float* bhh  = (const float*)d_in[10];
    const float* W1   = (const float*)d_in[11];
    const float* b1   = (const float*)d_in[12];
    const float* W2   = (const float*)d_in[13];
    const float* b2   = (const float*)d_in[14];
    const int* row = ei;          // edge_index[0]
    const int* col = ei + E;      // edge_index[1]

    // workspace carve (256B aligned)
    char* ws = (char*)d_ws;
    size_t off = 0;
    auto carve = [&](size_t bytes) { void* p = ws + off; off = (off + bytes + 255) & ~(size_t)255; return p; };
    float* out   = (float*)carve((size_t)N * 64 * 4);
    float* hbuf  = (float*)carve((size_t)N * 64 * 4);
    float* agg   = (float*)carve((size_t)N * 64 * 4);
    float* dinv  = (float*)carve((size_t)N * 4);
    float* normv = (float*)carve((size_t)E * 4);
    float* e_arr = (float*)carve((size_t)N * 4);
    float* a_num = (float*)carve((size_t)N * 4);
    float* h_l   = (float*)carve((size_t)B * 64 * 4);
    float* c_l   = (float*)carve((size_t)B * 64 * 4);
    float* rbuf  = (float*)carve((size_t)B * 64 * 4);
    float* qstar = (float*)carve((size_t)B * 128 * 4);
    unsigned int* m_u = (unsigned int*)carve((size_t)B * 4);
    float* denom = (float*)carve((size_t)B * 4);
    (void)ws_size; (void)in_sizes; (void)n_in; (void)out_size;

    float* y = (float*)d_out;

    // --- normalization coefficients ---
    deg_init_k<<<(N + 255) / 256, 256, 0, stream>>>(dinv, N);
    deg_edge_k<<<(E + 255) / 256, 256, 0, stream>>>(col, dinv, E);
    dinv_k<<<(N + 255) / 256, 256, 0, stream>>>(dinv, N);
    norm_k<<<(E + 255) / 256, 256, 0, stream>>>(row, col, dinv, normv, E);

    // --- input transform ---
    xform_k<<<N, 64, 0, stream>>>(x, W0, b0, out);

    // --- 6 propagation steps ---
    const int ntiles = N / 16;                       // 6250
    const int gemmBlocks = (ntiles + 7) / 8;
    const int totalND = N * 64;
    for (int s = 0; s < 6; ++s) {
        gemm_wc_k<<<gemmBlocks, 256, 0, stream>>>(out, Wc, hbuf, ntiles);
        agg_init_k<<<(totalND + 255) / 256, 256, 0, stream>>>(hbuf, dinv, agg, totalND);
        edge_msg_k<<<(E + 7) / 8, 256, 0, stream>>>(row, col, normv, hbuf, agg, E);
        relu_bias_k<<<(totalND + 255) / 256, 256, 0, stream>>>(agg, bc, out, totalND);
    }

    // --- Set2Set ---
    zero_f_k<<<(B * 64 + 255) / 256, 256, 0, stream>>>(h_l, B * 64);
    zero_f_k<<<(B * 64 + 255) / 256, 256, 0, stream>>>(c_l, B * 64);
    zero_f_k<<<(B * 128 + 255) / 256, 256, 0, stream>>>(qstar, B * 128);
    for (int s = 0; s < 6; ++s) {
        lstm_k<<<B, 256, 0, stream>>>(qstar, h_l, c_l, Wih, Whh, bih, bhh);
        att_init_k<<<B, 64, 0, stream>>>(m_u, denom, rbuf);
        e_max_k<<<(N + 7) / 8, 256, 0, stream>>>(out, h_l, bat, e_arr, m_u, N);
        anum_k<<<(N + 255) / 256, 256, 0, stream>>>(e_arr, bat, m_u, a_num, denom, N);
        r_accum_k<<<(N + 7) / 8, 256, 0, stream>>>(out, a_num, denom, bat, rbuf, N);
        qstar_k<<<B, 64, 0, stream>>>(h_l, rbuf, qstar);
    }

    // --- output MLP ---
    mlp_k<<<B, 64, 0, stream>>>(qstar, W1, b1, W2, b2, y);
}